// SoftSkeletonRecallLoss_9251359555893
// MI455X (gfx1250) — compile-verified
//
#include <hip/hip_runtime.h>
#include <stdint.h>

typedef __attribute__((ext_vector_type(2))) float v2f;
typedef __attribute__((ext_vector_type(8))) float v8f;

#define H_ 512
#define W_ 512
#define WPR 16            // 32-bit words per row
#define NWORDS (H_ * WPR) // 8192 words = 32KB per image

#if defined(__has_builtin)
#if __has_builtin(__builtin_amdgcn_wmma_f32_16x16x4_f32)
#define HAVE_WMMA_F32X4 1
#endif
#endif

// ---- bit-sliced sum of 8 one-bit values -> 4 bit-planes s0..s3 ----
__device__ __forceinline__ void sum8(uint32_t e0, uint32_t e1, uint32_t e2, uint32_t e3,
                                     uint32_t e4, uint32_t e5, uint32_t e6, uint32_t e7,
                                     uint32_t& s0, uint32_t& s1, uint32_t& s2, uint32_t& s3) {
    uint32_t xa = e0 ^ e1;
    uint32_t sa = xa ^ e2, ca = (e0 & e1) | (e2 & xa);
    uint32_t xb = e3 ^ e4;
    uint32_t sb = xb ^ e5, cb = (e3 & e4) | (e5 & xb);
    uint32_t sc = e6 ^ e7, cc = e6 & e7;
    uint32_t xs = sa ^ sb;
    s0 = xs ^ sc;
    uint32_t c0 = (sa & sb) | (sc & xs);
    uint32_t xt = ca ^ cb;
    uint32_t ts = xt ^ cc, tc = (ca & cb) | (cc & xt);
    s1 = ts ^ c0;
    uint32_t uc = ts & c0;
    s2 = tc ^ uc;
    s3 = tc & uc;
}

// One Zhang-Suen sub-iteration for row r; new words -> out[16]; returns "any pixel removed"
template <bool FIRST>
__device__ __forceinline__ bool zs_row(const uint32_t* img, int r, uint32_t* out) {
    const bool hasN = (r > 0), hasS = (r < H_ - 1);
    const uint32_t* rowN = img + (r - 1) * WPR;
    const uint32_t* rowC = img + r * WPR;
    const uint32_t* rowS = img + (r + 1) * WPR;
    uint32_t n0 = 0, c0 = 0, s0w = 0;
    uint32_t n1 = hasN ? rowN[0] : 0u, c1 = rowC[0], s1w = hasS ? rowS[0] : 0u;
    uint32_t n2 = hasN ? rowN[1] : 0u, c2 = rowC[1], s2w = hasS ? rowS[1] : 0u;
    bool changed = false;
#pragma unroll 4
    for (int w = 0; w < WPR; ++w) {
        uint32_t P2 = n1;
        uint32_t P3 = (n1 >> 1) | (n2 << 31);
        uint32_t P4 = (c1 >> 1) | (c2 << 31);
        uint32_t P5 = (s1w >> 1) | (s2w << 31);
        uint32_t P6 = s1w;
        uint32_t P7 = (s1w << 1) | (s0w >> 31);
        uint32_t P8 = (c1 << 1) | (c0 >> 31);
        uint32_t P9 = (n1 << 1) | (n0 >> 31);

        uint32_t b0, b1, b2b, b3;
        sum8(P2, P3, P4, P5, P6, P7, P8, P9, b0, b1, b2b, b3);
        (void)b3;
        uint32_t condB = (b1 | b2b) & ~(b0 & b1 & b2b); // 2 <= B <= 6

        uint32_t t0 = ~P2 & P3, t1 = ~P3 & P4, t2 = ~P4 & P5, t3 = ~P5 & P6;
        uint32_t t4 = ~P6 & P7, t5 = ~P7 & P8, t6 = ~P8 & P9, t7 = ~P9 & P2;
        uint32_t a0, a1, a2b, a3;
        sum8(t0, t1, t2, t3, t4, t5, t6, t7, a0, a1, a2b, a3);
        uint32_t A1 = a0 & ~a1 & ~a2b & ~a3; // A == 1

        uint32_t cond = FIRST ? (~(P2 & P4 & P6) & ~(P4 & P6 & P8))
                              : (~(P2 & P4 & P8) & ~(P2 & P6 & P8));

        uint32_t remove = c1 & condB & A1 & cond;
        out[w] = c1 & ~remove;
        changed |= (remove != 0u);

        n0 = n1; n1 = n2;
        c0 = c1; c1 = c2;
        s0w = s1w; s1w = s2w;
        const bool more = (w + 2 < WPR);
        n2 = (more && hasN) ? rowN[w + 2] : 0u;
        c2 = more ? rowC[w + 2] : 0u;
        s2w = (more && hasS) ? rowS[w + 2] : 0u;
    }
    return changed;
}

// Cross dilation for row r -> out[16]
__device__ __forceinline__ void dil_row(const uint32_t* img, int r, uint32_t* out) {
    const bool hasN = (r > 0), hasS = (r < H_ - 1);
    const uint32_t* rowN = img + (r - 1) * WPR;
    const uint32_t* rowC = img + r * WPR;
    const uint32_t* rowS = img + (r + 1) * WPR;
    uint32_t c0 = 0;
    uint32_t c1 = rowC[0];
    uint32_t c2 = rowC[1];
#pragma unroll 4
    for (int w = 0; w < WPR; ++w) {
        uint32_t nn = hasN ? rowN[w] : 0u;
        uint32_t ss = hasS ? rowS[w] : 0u;
        uint32_t e = (c1 >> 1) | (c2 << 31);  // im[c+1]
        uint32_t wv = (c1 << 1) | (c0 >> 31); // im[c-1]
        out[w] = c1 | nn | ss | e | wv;
        c0 = c1; c1 = c2;
        c2 = (w + 2 < WPR) ? rowC[w + 2] : 0u;
    }
}

// 16-word row writeback as 4x ds_store_b128
__device__ __forceinline__ void store_row(uint32_t* img, int r, const uint32_t* out) {
    uint4* dst = reinterpret_cast<uint4*>(img + r * WPR);
    const uint4* src = reinterpret_cast<const uint4*>(out);
#pragma unroll
    for (int q = 0; q < 4; ++q) dst[q] = src[q];
}

__global__ void __launch_bounds__(512) skel_recall_kernel(const float* __restrict__ x,
                                                          const int* __restrict__ y,
                                                          float* __restrict__ ws) {
    __shared__ uint32_t img[NWORDS]; // whole bit-packed image: 32KB
    __shared__ int flag;
    __shared__ float sInter, sGt;

    const int b = blockIdx.x;
    const int tid = threadIdx.x;
    const int r = tid; // one row per thread (blockDim == 512)

    // ---- binarize y -> bit-packed LDS ----
    {
        const int4* yrow = reinterpret_cast<const int4*>(y + (size_t)b * (H_ * W_) + (size_t)r * W_);
        uint32_t rowbits[WPR];
        for (int w = 0; w < WPR; ++w) {
            uint32_t bits = 0u;
#pragma unroll
            for (int q = 0; q < 8; ++q) {
                int4 v = yrow[w * 8 + q];
                bits |= (v.x > 0 ? 1u : 0u) << (q * 4 + 0);
                bits |= (v.y > 0 ? 1u : 0u) << (q * 4 + 1);
                bits |= (v.z > 0 ? 1u : 0u) << (q * 4 + 2);
                bits |= (v.w > 0 ? 1u : 0u) << (q * 4 + 3);
            }
            rowbits[w] = bits;
        }
        store_row(img, r, rowbits);
    }
    if (tid == 0) { flag = 1; sInter = 0.0f; sGt = 0.0f; }
    __syncthreads();

    // ---- Zhang-Suen thinning to convergence (all in LDS) ----
    uint32_t out[WPR];
    for (int guard = 0; guard < 4096; ++guard) {
        const bool run = (flag != 0);
        __syncthreads();
        if (!run) break;
        if (tid == 0) flag = 0;
        __syncthreads();

        bool ch = zs_row<true>(img, r, out);
        __syncthreads();
        store_row(img, r, out);
        if (ch) atomicOr(&flag, 1);
        __syncthreads();

        ch = zs_row<false>(img, r, out);
        __syncthreads();
        store_row(img, r, out);
        if (ch) atomicOr(&flag, 1);
        __syncthreads();
    }

    // ---- two cross dilations ----
    for (int d = 0; d < 2; ++d) {
        dil_row(img, r, out);
        __syncthreads();
        store_row(img, r, out);
        __syncthreads();
    }

    // ---- reduction: intersection = sum(x * skel), gt = popcount(skel) ----
    // WMMA mapping: chunk of 64 consecutive pixels -> A[m][k] = x[4m+k] (16x4 f32),
    // B[k][m] = bit[4m+k] (4x16 f32); accumulated diagonal D[m][m] = sum x*b.
    // wv forced scalar so the WMMA loop uses s_cbranch (EXEC stays all-ones).
    const int wv = __builtin_amdgcn_readfirstlane(tid >> 5);
    const int lane = tid & 31;
    const float* xb = x + (size_t)b * (H_ * W_);

    v8f acc = {};
    float gtp = 0.0f;
    float fInterFallback = 0.0f;

    for (int c = wv; c < (H_ * (W_ / 64)); c += 16) { // 4096 chunks of 64 pixels
        const int row = c >> 3;
        const int m = lane & 15;
        const int col = ((c & 7) << 6) + (m << 2);
        const float4 xv = *reinterpret_cast<const float4*>(xb + (size_t)row * W_ + col);
        const uint32_t nib = (img[row * WPR + (col >> 5)] >> (col & 31)) & 0xFu;

        const bool hi = (lane >= 16);
        const float ax = hi ? xv.z : xv.x;
        const float ay = hi ? xv.w : xv.y;
        const uint32_t bb = hi ? (nib >> 2) : nib;
        const float bx = (float)(bb & 1u);
        const float by = (float)((bb >> 1) & 1u);
        gtp += (float)__popc(bb & 3u);

        v2f a2 = {ax, ay};
        v2f b2 = {bx, by};
#ifdef HAVE_WMMA_F32X4
        acc = __builtin_amdgcn_wmma_f32_16x16x4_f32(false, a2, false, b2,
                                                    (short)0, acc, false, false);
#else
        fInterFallback += ax * bx + ay * by;
#endif
    }

    float dsum;
#ifdef HAVE_WMMA_F32X4
    // diagonal D[m][m]: m=0..7 in lane m (vgpr m); m=8..15 in lane m+16 (vgpr m-8)
    dsum = (lane < 8) ? acc[lane] : ((lane >= 24) ? acc[lane - 24] : 0.0f);
#else
    dsum = fInterFallback;
#endif
    for (int off = 16; off > 0; off >>= 1) {
        dsum += __shfl_down(dsum, off, 32);
        gtp += __shfl_down(gtp, off, 32);
    }
    if (lane == 0) {
        atomicAdd(&sInter, dsum);
        atomicAdd(&sGt, gtp);
    }
    __syncthreads();
    if (tid == 0) {
        ws[2 * b + 0] = sInter;
        ws[2 * b + 1] = sGt;
    }
}

__global__ void finalize_kernel(const float* __restrict__ ws, float* __restrict__ out, int nimg) {
    const int t = threadIdx.x;
    float rsum = 0.0f;
    if (t < nimg) rsum = (ws[2 * t + 0] + 1.0f) / (ws[2 * t + 1] + 1.0f);
    for (int off = 16; off > 0; off >>= 1) rsum += __shfl_down(rsum, off, 32);
    if (t == 0) out[0] = -rsum / (float)nimg;
}

extern "C" void kernel_launch(void* const* d_in, const int* in_sizes, int n_in,
                              void* d_out, int out_size, void* d_ws, size_t ws_size,
                              hipStream_t stream) {
    const float* x = (const float*)d_in[0];
    const int* y = (const int*)d_in[1];
    const int nimg = in_sizes[1] / (H_ * W_); // B*C images
    float* ws = (float*)d_ws;

    skel_recall_kernel<<<nimg, 512, 0, stream>>>(x, y, ws);
    finalize_kernel<<<1, 32, 0, stream>>>(ws, (float*)d_out, nimg);
}